// CopyEnhancedCodeT5_46437186404877
// MI455X (gfx1250) — compile-verified
//
#include <hip/hip_runtime.h>
#include <hip/hip_bf16.h>
#include <math.h>

typedef __attribute__((ext_vector_type(2))) float v2f;
typedef __attribute__((ext_vector_type(8))) float v8f;

// Problem dims (compile-time, from reference)
constexpr int B = 4, S = 512, T = 256, D = 1024, H = 16, V = 32105;
constexpr int TS = T * S;          // 131072
constexpr int HTS = H * T * S;     // 2097152
constexpr int SD = S * D;          // 524288
constexpr int TD = T * D;          // 262144
constexpr int BT = B * T;          // 1024
constexpr int BTS = B * T * S;     // 524288

// ---------------------------------------------------------------------------
// Kernel 1: A[b,t,s] = mean over heads of cross_attentions[b,h,t,s]
// ---------------------------------------------------------------------------
__global__ void head_mean_kernel(const float* __restrict__ ca,
                                 float* __restrict__ A) {
    int i = blockIdx.x * blockDim.x + threadIdx.x;   // over B*T*S
    if (i >= BTS) return;
    int b  = i / TS;
    int ts = i - b * TS;
    const float* p = ca + (size_t)b * HTS + ts;
    float acc = 0.f;
#pragma unroll
    for (int h = 0; h < H; ++h) acc += p[(size_t)h * TS];
    A[i] = acc * (1.0f / (float)H);
}

// ---------------------------------------------------------------------------
// Kernel 2: h_star[b,t,d] = sum_s A[b,t,s] * enc[b,s,d]  via V_WMMA_F32_16X16X4_F32
// One wave computes 4 T-tiles x 1 D-tile (64x16 output), so the strided
// B fragment is loaded once per K-step and reused by 4 WMMAs.
// Fragment layouts per CDNA5 ISA 7.12.2:
//   A 16x4  : lane L<16 -> row=L, {K0,K1}; lane L>=16 -> row=L-16, {K2,K3}
//   B 4x16  : lane L<16 -> col=L, {K0,K1}; lane L>=16 -> col=L-16, {K2,K3}
//   C 16x16 : vgpr r, lane L -> row = r + 8*(L>=16), col = L&15
// ---------------------------------------------------------------------------
__global__ void wmma_gemm_kernel(const float* __restrict__ A,
                                 const float* __restrict__ E,
                                 float* __restrict__ Hs) {
    constexpr int dTiles  = D / 16;   // 64
    constexpr int tGroups = T / 64;   // 4 groups of 4 tiles
    int wave = (blockIdx.x * blockDim.x + threadIdx.x) >> 5;   // 0..B*tGroups*dTiles-1
    int lane = threadIdx.x & 31;
    int b      = wave / (tGroups * dTiles);
    int rem    = wave - b * (tGroups * dTiles);
    int tGroup = rem / dTiles;                 // dTile fastest-varying: adjacent
    int dTile  = rem - tGroup * dTiles;        // waves load adjacent E columns

    const float* Ab = A + (size_t)b * TS;
    const float* Eb = E + (size_t)b * SD;

    int half = lane >> 4;            // 0 or 1
    int l    = lane & 15;
    int col  = dTile * 16 + l;       // B col (D index)
    int row0 = tGroup * 64 + l;      // first A row (T index) for this lane

    v8f c0 = {}, c1 = {}, c2 = {}, c3 = {};
    for (int k0 = 0; k0 < S; k0 += 4) {
        int ka = k0 + half * 2;
        // B fragment (column-strided): loaded once, reused by 4 WMMAs
        v2f bf;
        bf.x = Eb[(size_t)ka * D + col];
        bf.y = Eb[(size_t)(ka + 1) * D + col];
        // 4 A fragments: contiguous b64 loads
        const float* ap = Ab + (size_t)row0 * S + ka;
        v2f a0 = *(const v2f*)(ap);
        v2f a1 = *(const v2f*)(ap + 16 * S);
        v2f a2 = *(const v2f*)(ap + 32 * S);
        v2f a3 = *(const v2f*)(ap + 48 * S);
        c0 = __builtin_amdgcn_wmma_f32_16x16x4_f32(false, a0, false, bf, (short)0, c0, false, false);
        c1 = __builtin_amdgcn_wmma_f32_16x16x4_f32(false, a1, false, bf, (short)0, c1, false, false);
        c2 = __builtin_amdgcn_wmma_f32_16x16x4_f32(false, a2, false, bf, (short)0, c2, false, false);
        c3 = __builtin_amdgcn_wmma_f32_16x16x4_f32(false, a3, false, bf, (short)0, c3, false, false);
    }

    float* out = Hs + (size_t)b * TD + (size_t)(tGroup * 64 + half * 8) * D + col;
#pragma unroll
    for (int r = 0; r < 8; ++r) out[(size_t)r * D]        = c0[r];
#pragma unroll
    for (int r = 0; r < 8; ++r) out[(size_t)(r + 16) * D] = c1[r];
#pragma unroll
    for (int r = 0; r < 8; ++r) out[(size_t)(r + 32) * D] = c2[r];
#pragma unroll
    for (int r = 0; r < 8; ++r) out[(size_t)(r + 48) * D] = c3[r];
}

// ---------------------------------------------------------------------------
// Kernel 3: p_gen[b,t] = sigmoid(h_star . W[0:D] + dec . W[D:2D] + bias)
// One wave per (b,t).
// ---------------------------------------------------------------------------
__global__ void pgen_kernel(const float* __restrict__ Hs,
                            const float* __restrict__ Dec,
                            const float* __restrict__ W,     // [2D]
                            const float* __restrict__ bias,  // [1]
                            float* __restrict__ pgen) {
    int wave = (blockIdx.x * blockDim.x + threadIdx.x) >> 5;  // 0..BT-1
    int lane = threadIdx.x & 31;
    if (wave >= BT) return;
    const float* h = Hs  + (size_t)wave * D;
    const float* d = Dec + (size_t)wave * D;
    float acc = 0.f;
    for (int i = lane; i < D; i += 32)
        acc = fmaf(h[i], W[i], fmaf(d[i], W[D + i], acc));
#pragma unroll
    for (int off = 16; off > 0; off >>= 1)
        acc += __shfl_down(acc, off, 32);
    if (lane == 0)
        pgen[wave] = 1.f / (1.f + __expf(-(acc + bias[0])));
}

// ---------------------------------------------------------------------------
// Kernel 4: online softmax stats per row (single pass over logits)
// ---------------------------------------------------------------------------
__global__ void softmax_stats_kernel(const float* __restrict__ logits,
                                     float* __restrict__ rowmax,
                                     float* __restrict__ rowsum) {
    int row = blockIdx.x;                  // 0..BT-1
    int tid = threadIdx.x;
    const float* x = logits + (size_t)row * V;
    float m = -INFINITY, s = 0.f;
    for (int i = tid; i < V; i += blockDim.x) {
        float v  = x[i];
        float nm = fmaxf(m, v);
        s = s * __expf(m - nm) + __expf(v - nm);
        m = nm;
    }
    __shared__ float sm[256], ss[256];
    sm[tid] = m; ss[tid] = s;
    __syncthreads();
    for (int off = 128; off > 0; off >>= 1) {
        if (tid < off) {
            float m2 = sm[tid + off], s2 = ss[tid + off];
            float nm = fmaxf(sm[tid], m2);
            ss[tid] = ss[tid] * __expf(sm[tid] - nm) + s2 * __expf(m2 - nm);
            sm[tid] = nm;
        }
        __syncthreads();
    }
    if (tid == 0) { rowmax[row] = sm[0]; rowsum[row] = ss[0]; }
}

// ---------------------------------------------------------------------------
// Kernel 5: out[row, v] = p_gen[row] * exp(logits - max) / sum
// grid.y = row so no integer division on the hot path.
// ---------------------------------------------------------------------------
__global__ void combine_kernel(const float* __restrict__ logits,
                               const float* __restrict__ rowmax,
                               const float* __restrict__ rowsum,
                               const float* __restrict__ pgen,
                               float* __restrict__ out) {
    int row = blockIdx.y;
    int col = blockIdx.x * blockDim.x + threadIdx.x;
    if (col >= V) return;
    float scale = pgen[row] / rowsum[row];
    float m     = rowmax[row];
    size_t idx  = (size_t)row * V + col;
    out[idx] = scale * __expf(logits[idx] - m);
}

// ---------------------------------------------------------------------------
// Kernel 6: scatter copy-distribution:
//   out[b,t,ids[b,s]] += (1 - p_gen[b,t]) * A[b,t,s]   (fp32 HW atomic)
// ---------------------------------------------------------------------------
__global__ void scatter_kernel(const int* __restrict__ ids,
                               const float* __restrict__ A,
                               const float* __restrict__ pgen,
                               float* __restrict__ out) {
    int i = blockIdx.x * blockDim.x + threadIdx.x;   // over B*T*S
    if (i >= BTS) return;
    int b   = i / TS;
    int rem = i - b * TS;
    int t   = rem / S;
    int s   = rem - t * S;
    int v   = ids[b * S + s];
    int row = b * T + t;
    float w = (1.f - pgen[row]) * A[i];
    atomicAdd(out + (size_t)row * V + v, w);
}

// ---------------------------------------------------------------------------
extern "C" void kernel_launch(void* const* d_in, const int* in_sizes, int n_in,
                              void* d_out, int out_size, void* d_ws, size_t ws_size,
                              hipStream_t stream) {
    const int*   input_ids = (const int*)  d_in[0];
    const float* logits    = (const float*)d_in[1];
    const float* enc_h     = (const float*)d_in[2];
    const float* dec_h     = (const float*)d_in[3];
    const float* cross_att = (const float*)d_in[4];
    const float* W_gen_w   = (const float*)d_in[5];
    const float* W_gen_b   = (const float*)d_in[6];
    float* out = (float*)d_out;

    // Workspace layout (floats)
    char* ws = (char*)d_ws;
    float* A      = (float*)ws;                                   // B*T*S
    float* Hs     = (float*)(ws + (size_t)BTS * 4);               // B*T*D
    float* pgen   = (float*)(ws + (size_t)(BTS + B * TD) * 4);    // B*T
    float* rowmax = pgen + BT;                                    // B*T
    float* rowsum = rowmax + BT;                                  // B*T

    // 1) head mean
    head_mean_kernel<<<(BTS + 255) / 256, 256, 0, stream>>>(cross_att, A);

    // 2) WMMA fp32 GEMM: each wave does a 64(T) x 16(D) strip
    {
        int waves  = B * (T / 64) * (D / 16);   // 1024
        int blocks = waves / 4;                 // 128 threads/block (4 waves)
        wmma_gemm_kernel<<<blocks, 128, 0, stream>>>(A, enc_h, Hs);
    }

    // 3) p_gen: one wave per (b,t)
    pgen_kernel<<<(BT * 32 + 127) / 128, 128, 0, stream>>>(Hs, dec_h, W_gen_w, W_gen_b, pgen);

    // 4) softmax stats: one block per row
    softmax_stats_kernel<<<BT, 256, 0, stream>>>(logits, rowmax, rowsum);

    // 5) combine -> out
    {
        dim3 grid((V + 255) / 256, BT);
        combine_kernel<<<grid, 256, 0, stream>>>(logits, rowmax, rowsum, pgen, out);
    }

    // 6) scatter copy distribution (after combine, same stream)
    scatter_kernel<<<(BTS + 255) / 256, 256, 0, stream>>>(input_ids, A, pgen, out);
}